// LeGNN4_19567871000717
// MI455X (gfx1250) — compile-verified
//
#include <hip/hip_runtime.h>
#include <hip/hip_bf16.h>

typedef __attribute__((ext_vector_type(2))) float v2f;
typedef __attribute__((ext_vector_type(8))) float v8f;

// ---------------------------------------------------------------- zero fill
__global__ void zero_kernel(float* __restrict__ p, size_t n) {
    size_t i = (size_t)blockIdx.x * blockDim.x + threadIdx.x;
    size_t stride = (size_t)gridDim.x * blockDim.x;
    for (; i < n; i += stride) p[i] = 0.0f;
}

// ------------------------------------------- edge scatter: one wave per edge
// agg[dst] += h_src[src] (128 f32), ftsum[dst] += ft[e] (16 f32), cnt[dst] += 1
__global__ void scatter_kernel(const float* __restrict__ hsrc,
                               const float* __restrict__ ft,
                               const int*   __restrict__ src,
                               const int*   __restrict__ dst,
                               int E,
                               float* __restrict__ agg,
                               float* __restrict__ ftsum,
                               float* __restrict__ cnt) {
    int wave = (int)(((size_t)blockIdx.x * blockDim.x + threadIdx.x) >> 5);
    int lane = threadIdx.x & 31;
    if (wave >= E) return;
    int s = src[wave];
    int d = dst[wave];
    const float4 v = *reinterpret_cast<const float4*>(hsrc + (size_t)s * 128 + lane * 4);
    float* ap = agg + (size_t)d * 128 + lane * 4;
    unsafeAtomicAdd(ap + 0, v.x);
    unsafeAtomicAdd(ap + 1, v.y);
    unsafeAtomicAdd(ap + 2, v.z);
    unsafeAtomicAdd(ap + 3, v.w);
    if (lane < 16) unsafeAtomicAdd(ftsum + (size_t)d * 16 + lane, ft[(size_t)wave * 16 + lane]);
    if (lane == 0) unsafeAtomicAdd(cnt + d, 1.0f);
}

// --------------------------------------------------------- WMMA conv GEMM
// Out[m,:] (+)= H[m,:]@W1 + AggMean[m,:]@W2 + conv_b + FtMean[m,:]@emlpW + [cnt>0]*emlp_b
// W is (256x128) row-major: rows 0..127 = W1, rows 128..255 = W2.
// Block: 64 dst rows (4 m-tiles); 8 waves, wave w owns output cols [16w,16w+16).
// Each wave holds 4 accumulators; one B fragment feeds 4 WMMAs (B reuse 4x).
__global__ __launch_bounds__(256) void conv_gemm_kernel(
    const float* __restrict__ Hdst,   // [Ndst,128]
    const float* __restrict__ Agg,    // [Ndst,128] (segment sums)
    const float* __restrict__ FtSum,  // [Ndst,16]
    const float* __restrict__ Cnt,    // [Ndst]
    const float* __restrict__ W,      // [256,128]
    const float* __restrict__ convb,  // [128]
    const float* __restrict__ emlpW,  // [16,128]
    const float* __restrict__ emlpb,  // [128]
    float* __restrict__ Out,          // [Ndst,128]
    int Ndst, int accumulate)
{
    __shared__ float sA[64][132];   // dst node features (pad: bank = 4*row+col)
    __shared__ float sG[64][132];   // agg mean
    __shared__ float sF[64][18];    // ft mean (even pad keeps b64 aligned)
    __shared__ float sInv[64];
    __shared__ float sFlag[64];

    const int tid = threadIdx.x;
    const int m0  = blockIdx.x * 64;

    if (tid < 64) {
        int gm = m0 + tid;
        float c = (gm < Ndst) ? Cnt[gm] : 0.0f;
        sInv[tid]  = 1.0f / fmaxf(c, 1.0f);
        sFlag[tid] = (c > 0.0f) ? 1.0f : 0.0f;
    }
    __syncthreads();

    // stage 64x128 H and scaled Agg tiles as float4 (2048 vec4 each; 8/thread)
#pragma unroll
    for (int i = 0; i < 8; ++i) {
        int idx = tid + i * 256;          // float4 index
        int r = idx >> 5, c4 = (idx & 31) * 4;
        int gm = m0 + r;
        float4 a = make_float4(0.f, 0.f, 0.f, 0.f);
        float4 g = make_float4(0.f, 0.f, 0.f, 0.f);
        if (gm < Ndst) {
            a = *reinterpret_cast<const float4*>(Hdst + (size_t)gm * 128 + c4);
            g = *reinterpret_cast<const float4*>(Agg  + (size_t)gm * 128 + c4);
            float inv = sInv[r];
            g.x *= inv; g.y *= inv; g.z *= inv; g.w *= inv;
        }
        *reinterpret_cast<float4*>(&sA[r][c4]) = a;
        *reinterpret_cast<float4*>(&sG[r][c4]) = g;
    }
    // ft mean tile: 64x16 (1024 elems; 4/thread)
#pragma unroll
    for (int i = 0; i < 4; ++i) {
        int idx = tid + i * 256;
        int r = idx >> 4, c = idx & 15;
        int gm = m0 + r;
        sF[r][c] = (gm < Ndst) ? FtSum[(size_t)gm * 16 + c] * sInv[r] : 0.0f;
    }
    __syncthreads();

    const int lane = tid & 31;
    const int w    = tid >> 5;      // wave id -> n tile
    const int n0   = w * 16;
    const int ni   = lane & 15;     // A: row-in-tile ; B/D: col n
    const int half = lane >> 4;     // lane-half selects K pair / row-half

    v8f acc0 = {}, acc1 = {}, acc2 = {}, acc3 = {};

#define K_LOOP(SMEM, WPTR, KMAX)                                              \
    _Pragma("unroll 4")                                                       \
    for (int k0 = 0; k0 < (KMAX); k0 += 4) {                                  \
        const int ka = k0 + 2 * half;                                         \
        v2f b;                                                                \
        b[0] = (WPTR)[(size_t)ka * 128 + n0 + ni];                            \
        b[1] = (WPTR)[(size_t)(ka + 1) * 128 + n0 + ni];                      \
        const float2 t0 = *reinterpret_cast<const float2*>(&(SMEM)[ni +  0][ka]); \
        const float2 t1 = *reinterpret_cast<const float2*>(&(SMEM)[ni + 16][ka]); \
        const float2 t2 = *reinterpret_cast<const float2*>(&(SMEM)[ni + 32][ka]); \
        const float2 t3 = *reinterpret_cast<const float2*>(&(SMEM)[ni + 48][ka]); \
        v2f a0, a1, a2, a3;                                                   \
        a0[0] = t0.x; a0[1] = t0.y;                                           \
        a1[0] = t1.x; a1[1] = t1.y;                                           \
        a2[0] = t2.x; a2[1] = t2.y;                                           \
        a3[0] = t3.x; a3[1] = t3.y;                                           \
        acc0 = __builtin_amdgcn_wmma_f32_16x16x4_f32(false, a0, false, b, (short)0, acc0, false, false); \
        acc1 = __builtin_amdgcn_wmma_f32_16x16x4_f32(false, a1, false, b, (short)0, acc1, false, false); \
        acc2 = __builtin_amdgcn_wmma_f32_16x16x4_f32(false, a2, false, b, (short)0, acc2, false, false); \
        acc3 = __builtin_amdgcn_wmma_f32_16x16x4_f32(false, a3, false, b, (short)0, acc3, false, false); \
    }

    K_LOOP(sA, W, 128)              // H @ W1
    K_LOOP(sG, W + 128 * 128, 128)  // AggMean @ W2
    K_LOOP(sF, emlpW, 16)           // FtMean @ emlpW
#undef K_LOOP

    const float cb = convb[n0 + ni];
    const float eb = emlpb[n0 + ni];

#define EPILOGUE(ACC, T)                                                      \
    _Pragma("unroll")                                                         \
    for (int r = 0; r < 8; ++r) {                                             \
        int mrow = (T) * 16 + r + 8 * half;   /* D striping: vgpr r = rows r, r+8 */ \
        int gm = m0 + mrow;                                                   \
        if (gm < Ndst) {                                                      \
            float v = (ACC)[r] + cb + sFlag[mrow] * eb;                       \
            float* p = Out + (size_t)gm * 128 + n0 + ni;                      \
            *p = accumulate ? (*p + v) : v;                                   \
        }                                                                     \
    }

    EPILOGUE(acc0, 0)
    EPILOGUE(acc1, 1)
    EPILOGUE(acc2, 2)
    EPILOGUE(acc3, 3)
#undef EPILOGUE
}

// --------------------------------------------------- LayerNorm: one wave/row
__global__ void layernorm_kernel(const float* __restrict__ X,
                                 const float* __restrict__ g,
                                 const float* __restrict__ b,
                                 float* __restrict__ Y, int N) {
    int wave = (int)(((size_t)blockIdx.x * blockDim.x + threadIdx.x) >> 5);
    int lane = threadIdx.x & 31;
    if (wave >= N) return;
    const float4 v = *reinterpret_cast<const float4*>(X + (size_t)wave * 128 + lane * 4);
    float s  = v.x + v.y + v.z + v.w;
    float ss = v.x * v.x + v.y * v.y + v.z * v.z + v.w * v.w;
#pragma unroll
    for (int off = 16; off >= 1; off >>= 1) {
        s  += __shfl_xor(s,  off, 32);
        ss += __shfl_xor(ss, off, 32);
    }
    float mean = s * (1.0f / 128.0f);
    float var  = ss * (1.0f / 128.0f) - mean * mean;
    float inv  = rsqrtf(var + 1e-5f);
    int c = lane * 4;
    float4 o;
    o.x = (v.x - mean) * inv * g[c + 0] + b[c + 0];
    o.y = (v.y - mean) * inv * g[c + 1] + b[c + 1];
    o.z = (v.z - mean) * inv * g[c + 2] + b[c + 2];
    o.w = (v.w - mean) * inv * g[c + 3] + b[c + 3];
    *reinterpret_cast<float4*>(Y + (size_t)wave * 128 + lane * 4) = o;
}

// ---------------------------------- h_dn gets no messages -> LN(0) = ln_b[3]
__global__ void fillrow_kernel(float* __restrict__ Y, const float* __restrict__ b, size_t n) {
    size_t i = (size_t)blockIdx.x * blockDim.x + threadIdx.x;
    if (i < n) Y[i] = b[i & 127];
}

extern "C" void kernel_launch(void* const* d_in, const int* in_sizes, int n_in,
                              void* d_out, int out_size, void* d_ws, size_t ws_size,
                              hipStream_t stream) {
    const size_t D = 128;
    const float* h_in[4]  = {(const float*)d_in[0], (const float*)d_in[1],
                             (const float*)d_in[2], (const float*)d_in[3]};
    const float* ftp[4]   = {(const float*)d_in[4], (const float*)d_in[5],
                             (const float*)d_in[6], (const float*)d_in[7]};
    const float* conv_W = (const float*)d_in[8];
    const float* conv_b = (const float*)d_in[9];
    const float* emlp_W = (const float*)d_in[10];
    const float* emlp_b = (const float*)d_in[11];
    const float* ln_g   = (const float*)d_in[12];
    const float* ln_b   = (const float*)d_in[13];
    const int* srcp[4]  = {(const int*)d_in[14], (const int*)d_in[16],
                           (const int*)d_in[18], (const int*)d_in[20]};
    const int* dstp[4]  = {(const int*)d_in[15], (const int*)d_in[17],
                           (const int*)d_in[19], (const int*)d_in[21]};
    int E[4] = {in_sizes[14], in_sizes[16], in_sizes[18], in_sizes[20]};

    size_t Nn[4] = {(size_t)in_sizes[0] / D, (size_t)in_sizes[1] / D,
                    (size_t)in_sizes[2] / D, (size_t)in_sizes[3] / D};
    const size_t NB = Nn[0], NL = Nn[1], NC = Nn[2], ND = Nn[3];
    const size_t NT = NB + NL + NC + ND;

    // edge meta: src node type, dst node type
    const int st[4] = {1, 2, 3, 0};
    const int dt[4] = {0, 0, 1, 2};
    const size_t dstN[4] = {NB, NB, NL, NC};

    // workspace layout (floats)
    float* ws = (float*)d_ws;
    float* hA = ws;                                  // NT*128 intermediate h
    float* hAp[4] = {hA, hA + NB * D, hA + (NB + NL) * D, hA + (NB + NL + NC) * D};
    float* outb = hA + NT * D;                       // (NB+NL+NC)*128 accumulators
    float* outp[3] = {outb, outb + NB * D, outb + (NB + NL) * D};
    float* aggb = outb + (NB + NL + NC) * D;
    float* aggp[4]; { size_t o = 0; for (int e = 0; e < 4; ++e) { aggp[e] = aggb + o; o += dstN[e] * D; } }
    size_t aggTot = (NB + NB + NL + NC) * D;
    float* ftsb = aggb + aggTot;
    float* ftsp[4]; { size_t o = 0; for (int e = 0; e < 4; ++e) { ftsp[e] = ftsb + o; o += dstN[e] * 16; } }
    size_t ftsTot = (NB + NB + NL + NC) * 16;
    float* cntb = ftsb + ftsTot;
    float* cntp[4]; { size_t o = 0; for (int e = 0; e < 4; ++e) { cntp[e] = cntb + o; o += dstN[e]; } }
    size_t cntTot = NB + NB + NL + NC;
    size_t zeroTot = aggTot + ftsTot + cntTot;       // contiguous agg|fts|cnt

    float* O = (float*)d_out;
    float* Op[4] = {O, O + NB * D, O + (NB + NL) * D, O + (NB + NL + NC) * D};

    for (int l = 0; l < 2; ++l) {
        const float* hcur[4];
        float* hnext[4];
        for (int nt = 0; nt < 4; ++nt) {
            hcur[nt]  = (l == 0) ? h_in[nt] : hAp[nt];
            hnext[nt] = (l == 0) ? hAp[nt]  : Op[nt];
        }

        zero_kernel<<<2048, 256, 0, stream>>>(aggb, zeroTot);

        for (int e = 0; e < 4; ++e) {
            int blocks = (E[e] + 7) / 8;   // 8 waves (edges) per 256-thread block
            scatter_kernel<<<blocks, 256, 0, stream>>>(
                hcur[st[e]], ftp[e], srcp[e], dstp[e], E[e],
                aggp[e], ftsp[e], cntp[e]);
        }

        for (int e = 0; e < 4; ++e) {
            int nt = dt[e];
            int Nd = (int)Nn[nt];
            int accum = (e == 1) ? 1 : 0;  // bv receives e0 (write) then e1 (accumulate)
            conv_gemm_kernel<<<(Nd + 63) / 64, 256, 0, stream>>>(
                hcur[nt], aggp[e], ftsp[e], cntp[e],
                conv_W + ((size_t)l * 4 + e) * 256 * D,
                conv_b + ((size_t)l * 4 + e) * D,
                emlp_W + (size_t)e * 16 * D,
                emlp_b + (size_t)e * D,
                outp[nt], Nd, accum);
        }

        for (int nt = 0; nt < 3; ++nt) {
            int Nr = (int)Nn[nt];
            layernorm_kernel<<<(Nr + 7) / 8, 256, 0, stream>>>(
                outp[nt], ln_g + nt * D, ln_b + nt * D, hnext[nt], Nr);
        }
        // donor nodes receive no messages: LN(0) = ln_b[3]
        {
            size_t n = ND * D;
            fillrow_kernel<<<(int)((n + 255) / 256), 256, 0, stream>>>(hnext[3], ln_b + 3 * D, n);
        }
    }
}